// GCN_31413390803465
// MI455X (gfx1250) — compile-verified
//
#include <hip/hip_runtime.h>
#include <hip/hip_bf16.h>
#include <math.h>

// ---------------------------------------------------------------------------
// GCN forward for MI455X (gfx1250, wave32).
//   layer: H = X @ W  (WMMA f32 16x16x4; W staged in LDS via async-to-LDS)
//          AGG = bias ; AGG[dst] += H[src] * ew  (atomic scatter, warp/edge)
//          X' = relu(AGG / max(||AGG||_2, eps))  (warp/node, shfl_xor reduce)
//   head:  logits = emb @ linW + linb ; probs = softmax(logits)
// ---------------------------------------------------------------------------

typedef __attribute__((ext_vector_type(2))) float v2f;
typedef __attribute__((ext_vector_type(8))) float v8f;
typedef __attribute__((ext_vector_type(4))) int   v4i;
typedef __attribute__((address_space(1))) v4i     gv4i;   // global v4i
typedef __attribute__((address_space(3))) v4i     lv4i;   // LDS v4i

#define GCN_EPS 1e-12f

#if __has_builtin(__builtin_amdgcn_global_load_async_to_lds_b128)
#define GCN_HAVE_ASYNC_LDS 1
#else
#define GCN_HAVE_ASYNC_LDS 0
#endif

// ---- WMMA fp32 GEMM: H[M,NOUT] = X[M,K] @ W[K,NOUT].
// One wave computes MT stacked 16x16 tiles (same N-tile), sharing the B
// fragment (read from LDS) across the MT accumulators each k-step. W (<=49KB)
// is staged once per workgroup with GLOBAL_LOAD_ASYNC_TO_LDS_B128 (ASYNCcnt).
// A-frag (16x4 f32, 2 VGPR): lanes 0-15 hold K={0,1}, lanes 16-31 K={2,3}.
// B-frag (4x16): VGPR0 = rows {K=0 | K=2}, VGPR1 = rows {K=1 | K=3}.
// C/D: VGPR r -> row r (lanes 0-15) / r+8 (lanes 16-31).
template <int K, int NOUT, int MT>
__global__ void gcn_gemm_wmma(const float* __restrict__ X,
                              const float* __restrict__ W,
                              float* __restrict__ H, int M) {
  constexpr int TILES_N = NOUT / 16;
  __shared__ float sW[K * NOUT];

  // ---- cooperative W staging (all waves participate, then barrier)
  const int tid = threadIdx.x;
#if GCN_HAVE_ASYNC_LDS
  {
    constexpr int GRANULES = (K * NOUT) / 4;     // 16B granules
    for (int i = tid; i < GRANULES; i += 256) {
      float* g = const_cast<float*>(W) + i * 4;
      float* l = sW + i * 4;
      __builtin_amdgcn_global_load_async_to_lds_b128(
          (gv4i*)g, (lv4i*)l, /*offset=*/0, /*cpol=*/0);
    }
#if __has_builtin(__builtin_amdgcn_s_wait_asynccnt)
    __builtin_amdgcn_s_wait_asynccnt(0);
#else
    asm volatile("s_wait_asynccnt 0" ::: "memory");
#endif
  }
#else
  for (int i = tid; i < K * NOUT; i += 256) sW[i] = W[i];
#endif
  __syncthreads();

  const int wave   = (blockIdx.x * blockDim.x + tid) >> 5;
  const int mblk   = wave / TILES_N;          // block of MT m-tiles
  const int tile_n = wave - mblk * TILES_N;
  const int row0   = mblk * (MT * 16);
  if (row0 >= M) return;                      // wave-uniform: EXEC all-1s

  const int lane = tid & 31;
  const int half = lane >> 4;                 // 0: lanes 0-15, 1: lanes 16-31
  const int sub  = lane & 15;
  const int bcol = tile_n * 16 + sub;

  const float* __restrict__ xr = X + (size_t)(row0 + sub) * K; // A row base
  const float* wl = sW + bcol;                                 // B col (LDS)

  v8f acc[MT];
#pragma unroll
  for (int t = 0; t < MT; ++t) acc[t] = (v8f){};

#pragma unroll
  for (int k = 0; k < K; k += 4) {
    const int ka = k + 2 * half;              // this lane's K pair (even)
    v2f b;
    b.x = wl[ka * NOUT];                      // ds_load_b32, imm offset
    b.y = wl[(ka + 1) * NOUT];
#pragma unroll
    for (int t = 0; t < MT; ++t) {
      v2f a;
      a.x = xr[t * 16 * K + ka];
      a.y = xr[t * 16 * K + ka + 1];
      acc[t] = __builtin_amdgcn_wmma_f32_16x16x4_f32(
          /*neg_a=*/false, a, /*neg_b=*/false, b,
          /*c_mod=*/(short)0, acc[t], /*reuse_a=*/false, /*reuse_b=*/false);
    }
  }

  float* __restrict__ hb = H + (size_t)(row0 + 8 * half) * NOUT + bcol;
#pragma unroll
  for (int t = 0; t < MT; ++t)
#pragma unroll
    for (int r = 0; r < 8; ++r)
      hb[(size_t)(t * 16 + r) * NOUT] = acc[t][r];
}

// ---- AGG[n][f] = bias[f]  (fuses the +b of the layer; scatter adds on top)
__global__ void gcn_bias_init(float* __restrict__ agg,
                              const float* __restrict__ bias, int N, int F) {
  int i = blockIdx.x * blockDim.x + threadIdx.x;
  if (i < N * F) agg[i] = bias[i % F];
}

// ---- warp per edge: AGG[dst] += H[src] * ew   (global_atomic_add_f32)
__global__ void gcn_scatter_add(const float* __restrict__ H,
                                const int* __restrict__ src,
                                const int* __restrict__ dst,
                                const float* __restrict__ ew,
                                float* __restrict__ agg, int E, int F) {
  const int e = (blockIdx.x * blockDim.x + threadIdx.x) >> 5;
  if (e >= E) return;
  const int lane = threadIdx.x & 31;
  const int s = src[e];
  const int d = dst[e];
  const float w = ew[e];
  const float* __restrict__ hs = H + (size_t)s * F;
  float* __restrict__ ad = agg + (size_t)d * F;
  for (int f = lane; f < F; f += 32)
    atomicAdd(&ad[f], hs[f] * w);
}

// ---- warp per node: out = relu(in / max(||in||, eps)); F multiple of 32
__global__ void gcn_l2norm_relu(const float* __restrict__ in,
                                float* __restrict__ out, int N, int F) {
  const int node = (blockIdx.x * blockDim.x + threadIdx.x) >> 5;
  if (node >= N) return;
  const int lane = threadIdx.x & 31;
  const float* __restrict__ xi = in + (size_t)node * F;
  float v[3];                               // F=96 -> 3 per lane
  float ss = 0.f;
  int c = 0;
  for (int f = lane; f < F; f += 32, ++c) {
    v[c] = xi[f];
    ss += v[c] * v[c];
  }
#pragma unroll
  for (int off = 16; off > 0; off >>= 1) ss += __shfl_xor(ss, off, 32);
  const float inv = 1.f / fmaxf(sqrtf(ss), GCN_EPS);
  float* __restrict__ xo = out + (size_t)node * F;
  c = 0;
  for (int f = lane; f < F; f += 32, ++c)
    xo[f] = fmaxf(v[c] * inv, 0.f);
}

// ---- thread per (node, class): logits = emb @ linW + linb
__global__ void gcn_logits(const float* __restrict__ emb,
                           const float* __restrict__ linW,
                           const float* __restrict__ linb,
                           float* __restrict__ logits, int N, int F, int C) {
  const int i = blockIdx.x * blockDim.x + threadIdx.x;
  if (i >= N * C) return;
  const int node = i / C;
  const int cls  = i - node * C;
  const float* __restrict__ e = emb + (size_t)node * F;
  float acc = linb[cls];
  for (int k = 0; k < F; ++k) acc += e[k] * linW[(size_t)k * C + cls];
  logits[i] = acc;
}

// ---- thread per node: stable softmax over C classes
__global__ void gcn_softmax(const float* __restrict__ logits,
                            float* __restrict__ probs, int N, int C) {
  const int node = blockIdx.x * blockDim.x + threadIdx.x;
  if (node >= N) return;
  const float* __restrict__ l = logits + (size_t)node * C;
  float* __restrict__ p = probs + (size_t)node * C;
  float m = l[0];
  for (int cc = 1; cc < C; ++cc) m = fmaxf(m, l[cc]);
  float s = 0.f;
  for (int cc = 0; cc < C; ++cc) {
    const float ev = __expf(l[cc] - m);
    p[cc] = ev;
    s += ev;
  }
  const float inv = 1.f / s;
  for (int cc = 0; cc < C; ++cc) p[cc] *= inv;
}

extern "C" void kernel_launch(void* const* d_in, const int* in_sizes, int n_in,
                              void* d_out, int out_size, void* d_ws, size_t ws_size,
                              hipStream_t stream) {
  const int N = 50000, E = 800000, C = 40;
  constexpr int F0 = 128, F = 96, MT = 5;    // 50000/16 = 3125 = 5 * 625

  const float* x    = (const float*)d_in[0];
  const int*   ei   = (const int*)d_in[1];     // [2, E] row-major
  const float* ew   = (const float*)d_in[2];
  const float* W1   = (const float*)d_in[3];
  const float* b1   = (const float*)d_in[4];
  const float* W2   = (const float*)d_in[5];
  const float* b2   = (const float*)d_in[6];
  const float* W3   = (const float*)d_in[7];
  const float* b3   = (const float*)d_in[8];
  const float* linW = (const float*)d_in[9];
  const float* linb = (const float*)d_in[10];
  const int* src = ei;
  const int* dst = ei + E;

  float* H  = (float*)d_ws;                    // [N, F] gemm output
  float* XA = H  + (size_t)N * F;              // [N, F] ping
  float* XB = XA + (size_t)N * F;              // [N, F] pong

  float* out    = (float*)d_out;
  float* emb    = out;                         // [N, 96]
  float* logits = emb + (size_t)N * F;         // [N, 40]
  float* probs  = logits + (size_t)N * C;      // [N, 40]

  const int mblks     = (N / 16 + MT - 1) / MT;       // 625
  const int gemmWaves = mblks * (F / 16);             // 3750
  const int gemmBlks  = (gemmWaves * 32 + 255) / 256;
  const int biasBlks  = (N * F + 255) / 256;
  const int scatBlks  = (E * 32 + 255) / 256;
  const int normBlks  = (N * 32 + 255) / 256;

  // ---- layer 1: x[N,128] -> XA[N,96]
  gcn_gemm_wmma<F0, F, MT><<<gemmBlks, 256, 0, stream>>>(x, W1, H, N);
  gcn_bias_init<<<biasBlks, 256, 0, stream>>>(XA, b1, N, F);
  gcn_scatter_add<<<scatBlks, 256, 0, stream>>>(H, src, dst, ew, XA, E, F);
  gcn_l2norm_relu<<<normBlks, 256, 0, stream>>>(XA, XA, N, F);

  // ---- layer 2: XA -> XB
  gcn_gemm_wmma<F, F, MT><<<gemmBlks, 256, 0, stream>>>(XA, W2, H, N);
  gcn_bias_init<<<biasBlks, 256, 0, stream>>>(XB, b2, N, F);
  gcn_scatter_add<<<scatBlks, 256, 0, stream>>>(H, src, dst, ew, XB, E, F);
  gcn_l2norm_relu<<<normBlks, 256, 0, stream>>>(XB, XB, N, F);

  // ---- layer 3: XB -> emb (normalize writes straight into d_out)
  gcn_gemm_wmma<F, F, MT><<<gemmBlks, 256, 0, stream>>>(XB, W3, H, N);
  gcn_bias_init<<<biasBlks, 256, 0, stream>>>(XA, b3, N, F);
  gcn_scatter_add<<<scatBlks, 256, 0, stream>>>(H, src, dst, ew, XA, E, F);
  gcn_l2norm_relu<<<normBlks, 256, 0, stream>>>(XA, emb, N, F);

  // ---- head
  gcn_logits<<<(N * C + 255) / 256, 256, 0, stream>>>(emb, linW, linb, logits, N, F, C);
  gcn_softmax<<<(N + 255) / 256, 256, 0, stream>>>(logits, probs, N, C);
}